// MyModel_87522843558620
// MI455X (gfx1250) — compile-verified
//
#include <hip/hip_runtime.h>
#include <hip/hip_bf16.h>

#define N_ANCHORS   109120
#define NUM_CLASSES 80
#define SCORE_THR   0.5f
#define IOU_THR     0.45f
#define MAX_BOXES   100
#define NBINS       64
#define BIN_SCALE   128.0f   /* NBINS / (1.0 - SCORE_THR) */
#define CAP         4096
#define TARGET      1280
#define PREF        512

#if defined(__has_builtin)
#if __has_builtin(__builtin_amdgcn_global_load_async_to_lds_b64) && __has_builtin(__builtin_amdgcn_s_wait_asynccnt)
#define HAVE_ASYNC_LDS 1
#endif
#endif
#ifndef HAVE_ASYNC_LDS
#define HAVE_ASYNC_LDS 0
#endif

#if HAVE_ASYNC_LDS
// Builtin signature (from clang diagnostic): takes pointers to int __vector(2),
// global (AS1) source and LDS (AS3) destination, plus imm offset and cpol.
typedef int v2i_t __attribute__((ext_vector_type(2)));
typedef __attribute__((address_space(1))) v2i_t* gv2i_p;
typedef __attribute__((address_space(3))) v2i_t* lv2i_p;
#endif

// ---- packed sort key: monotonic-in-score (scores are > 0), smaller index wins ties ----
__device__ __forceinline__ unsigned long long packkey(float s, int idx) {
  unsigned mk = __float_as_uint(s) | 0x80000000u;   // s >= 0.5 > 0 always when called
  return ((unsigned long long)mk << 32) | (unsigned)(~(unsigned)idx);
}
__device__ __forceinline__ float key_score(unsigned long long k) {
  return __uint_as_float((unsigned)(k >> 32) & 0x7FFFFFFFu);
}
__device__ __forceinline__ unsigned key_idx(unsigned long long k) {
  return ~(unsigned)k;
}

// IoU identical to reference math: inter / (areaA + areaB - inter), 0 if union<=0
__device__ __forceinline__ float iou2(const float4 a, float areaA, const float4 b) {
  float y1 = fmaxf(a.x, b.x);
  float x1 = fmaxf(a.y, b.y);
  float y2 = fminf(a.z, b.z);
  float x2 = fminf(a.w, b.w);
  float inter = fmaxf(y2 - y1, 0.0f) * fmaxf(x2 - x1, 0.0f);
  float areaB = (b.z - b.x) * (b.w - b.y);
  float uni = areaA + areaB - inter;
  return (uni > 0.0f) ? (inter / uni) : 0.0f;
}

// ---------------- kernel 1: boxes = anchors + pr ----------------
__global__ __launch_bounds__(256) void prep_boxes(const float4* __restrict__ anch,
                                                  const float4* __restrict__ pr,
                                                  float4* __restrict__ boxes) {
  int i = blockIdx.x * blockDim.x + threadIdx.x;
  if (i < N_ANCHORS) {
    float4 a = anch[i], p = pr[i];
    boxes[i] = make_float4(a.x + p.x, a.y + p.y, a.z + p.z, a.w + p.w);
  }
}

// ---------------- kernel 2: per-class score histogram ----------------
__global__ __launch_bounds__(256) void hist_kernel(const float* __restrict__ conf,
                                                   unsigned* __restrict__ ghist) {
  __shared__ unsigned h[NUM_CLASSES * NBINS];
  for (int t = threadIdx.x; t < NUM_CLASSES * NBINS; t += blockDim.x) h[t] = 0u;
  __syncthreads();
  int i = blockIdx.x * blockDim.x + threadIdx.x;
  if (i < N_ANCHORS) {
    const float4* row4 = (const float4*)(conf + (size_t)i * NUM_CLASSES);
    __builtin_prefetch(row4, 0, 3);                       // global_prefetch_b8
    __builtin_prefetch((const char*)row4 + 128, 0, 3);
    __builtin_prefetch((const char*)row4 + 256, 0, 3);
    for (int q = 0; q < NUM_CLASSES / 4; ++q) {
      float4 v = row4[q];
      float sv[4] = {v.x, v.y, v.z, v.w};
#pragma unroll
      for (int r = 0; r < 4; ++r) {
        float s = sv[r];
        if (s >= SCORE_THR) {
          int b = (int)((s - SCORE_THR) * BIN_SCALE);
          b = (b > NBINS - 1) ? NBINS - 1 : b;
          atomicAdd(&h[(q * 4 + r) * NBINS + b], 1u);     // ds_add_u32
        }
      }
    }
  }
  __syncthreads();
  for (int t = threadIdx.x; t < NUM_CLASSES * NBINS; t += blockDim.x) {
    unsigned v = h[t];
    if (v) atomicAdd(&ghist[t], v);
  }
}

// ---------------- kernel 3: per-class collection threshold ----------------
__global__ void thresh_kernel(const unsigned* __restrict__ ghist,
                              int* __restrict__ bsel,
                              unsigned* __restrict__ totals) {
  int c = blockIdx.x * blockDim.x + threadIdx.x;
  if (c < NUM_CLASSES) {
    unsigned acc = 0; int chosen = 0; bool found = false;
    for (int b = NBINS - 1; b >= 0; --b) {
      acc += ghist[c * NBINS + b];
      if (!found && acc >= (unsigned)TARGET) { chosen = b; found = true; }
    }
    bsel[c] = found ? chosen : 0;
    totals[c] = acc;   // count of scores >= SCORE_THR in class c
  }
}

// ---------------- kernel 4: collect top candidates per class ----------------
__global__ __launch_bounds__(256) void collect_kernel(const float* __restrict__ conf,
                                                      const int* __restrict__ bsel,
                                                      unsigned* __restrict__ cnt,
                                                      unsigned long long* __restrict__ cand) {
  __shared__ int sb[NUM_CLASSES];
  for (int t = threadIdx.x; t < NUM_CLASSES; t += blockDim.x) sb[t] = bsel[t];
  __syncthreads();
  int i = blockIdx.x * blockDim.x + threadIdx.x;
  if (i < N_ANCHORS) {
    const float4* row4 = (const float4*)(conf + (size_t)i * NUM_CLASSES);
    __builtin_prefetch(row4, 0, 3);
    __builtin_prefetch((const char*)row4 + 128, 0, 3);
    __builtin_prefetch((const char*)row4 + 256, 0, 3);
    for (int q = 0; q < NUM_CLASSES / 4; ++q) {
      float4 v = row4[q];
      float sv[4] = {v.x, v.y, v.z, v.w};
#pragma unroll
      for (int r = 0; r < 4; ++r) {
        float s = sv[r];
        if (s >= SCORE_THR) {
          int b = (int)((s - SCORE_THR) * BIN_SCALE);
          b = (b > NBINS - 1) ? NBINS - 1 : b;
          int c = q * 4 + r;
          if (b >= sb[c]) {
            unsigned pos = atomicAdd(&cnt[c], 1u);
            if (pos < (unsigned)CAP) cand[(size_t)c * CAP + pos] = packkey(s, i);
          }
        }
      }
    }
  }
}

// ---------------- kernel 5: per-class sort + greedy NMS scan ----------------
__global__ __launch_bounds__(1024) void nms_kernel(const float* __restrict__ conf,
                                                   const float4* __restrict__ boxes,
                                                   const unsigned* __restrict__ cnt,
                                                   const unsigned* __restrict__ totals,
                                                   const unsigned long long* __restrict__ cand,
                                                   float* __restrict__ out) {
  const int c = blockIdx.x;
  const int tid = (int)threadIdx.x;

  __shared__ unsigned long long keys[CAP];     // 32 KB
  __shared__ float4 cbox[PREF];                // 8 KB
  __shared__ float4 selbox[MAX_BOXES];
  __shared__ unsigned long long sred[32];
  __shared__ unsigned long long sh_best;
  __shared__ int sh_nsel;
  __shared__ int sh_done;

  float* out_sc = out;                               // [80][100]
  float* out_bx = out + NUM_CLASSES * MAX_BOXES;     // [80][100][4]
  float* out_id = out + NUM_CLASSES * MAX_BOXES * 5; // [80][100]

  // deterministic zero-fill of this class's output slots
  for (int t = tid; t < MAX_BOXES; t += blockDim.x) {
    out_sc[c * MAX_BOXES + t] = 0.0f;
    out_id[c * MAX_BOXES + t] = 0.0f;
  }
  for (int t = tid; t < MAX_BOXES * 4; t += blockDim.x)
    out_bx[c * MAX_BOXES * 4 + t] = 0.0f;

  unsigned rawcnt = cnt[c];
  bool overflow = rawcnt > (unsigned)CAP;      // statistically impossible; exact fallback below
  int n = overflow ? 0 : (int)rawcnt;

  for (int t = tid; t < CAP; t += blockDim.x)
    if (t >= n) keys[t] = 0ull;                // padding sorts last
#if HAVE_ASYNC_LDS
  for (int t = tid; t < n; t += blockDim.x) {
    __builtin_amdgcn_global_load_async_to_lds_b64(
        (gv2i_p)(cand + (size_t)c * CAP + t),
        (lv2i_p)&keys[t], 0, 0);
  }
  __builtin_amdgcn_s_wait_asynccnt(0);
#else
  for (int t = tid; t < n; t += blockDim.x)
    keys[t] = cand[(size_t)c * CAP + t];
#endif
  if (tid == 0) { sh_nsel = 0; sh_done = 0; }
  __syncthreads();

  // bitonic sort, descending, 4096 keys, 1024 threads x 4 elements
  for (int k = 2; k <= CAP; k <<= 1) {
    for (int j = k >> 1; j > 0; j >>= 1) {
      for (int t = tid; t < CAP; t += blockDim.x) {
        int p = t ^ j;
        if (p > t) {
          unsigned long long a = keys[t], b = keys[p];
          bool asc = (t & k) != 0;
          if (asc ? (a > b) : (a < b)) { keys[t] = b; keys[p] = a; }
        }
      }
      __syncthreads();
    }
  }

  // stage boxes of the leading candidates into LDS for the sequential scan
  int np = (n < PREF) ? n : PREF;
  for (int t = tid; t < np; t += blockDim.x)
    cbox[t] = boxes[key_idx(keys[t])];
  __syncthreads();

  // sequential greedy scan on wave 0: accept iff IoU<=thr vs every accepted box
  if (tid < 32) {
    int nsel = 0;
    for (int j = 0; j < n && nsel < MAX_BOXES; ++j) {
      unsigned long long k = keys[j];
      float4 bx = (j < PREF) ? cbox[j] : boxes[key_idx(k)];
      float areaA = (bx.z - bx.x) * (bx.w - bx.y);
      bool ov = false;
      for (int q = tid; q < nsel; q += 32)
        ov = ov || (iou2(bx, areaA, selbox[q]) > IOU_THR);
      if (__ballot(ov) == 0ull) {
        if (tid == 0) {
          selbox[nsel] = bx;
          out_sc[c * MAX_BOXES + nsel] = key_score(k);
          out_bx[(c * MAX_BOXES + nsel) * 4 + 0] = bx.x;
          out_bx[(c * MAX_BOXES + nsel) * 4 + 1] = bx.y;
          out_bx[(c * MAX_BOXES + nsel) * 4 + 2] = bx.z;
          out_bx[(c * MAX_BOXES + nsel) * 4 + 3] = bx.w;
          out_id[c * MAX_BOXES + nsel] = (float)(c + 1);
        }
        nsel++;
      }
    }
    if (tid == 0) {
      sh_nsel = nsel;
      sh_best = (n > 0) ? keys[n - 1] : ~0ull;  // last processed key in descending order
    }
  }

  // exact continuation (rare): resume descending traversal directly from conf
  unsigned total = totals[c];
  bool more = overflow ? (total > 0u) : (total > (unsigned)n);
  if (more) {
    for (;;) {
      __syncthreads();
      if (sh_done || sh_nsel >= MAX_BOXES) break;
      unsigned long long lastKey = sh_best;
      unsigned long long best = 0ull;
      for (int i = tid; i < N_ANCHORS; i += blockDim.x) {
        float s = conf[(size_t)i * NUM_CLASSES + c];
        if (s >= SCORE_THR) {
          unsigned long long k2 = packkey(s, i);
          if (k2 < lastKey && k2 > best) best = k2;
        }
      }
      for (int off = 16; off > 0; off >>= 1) {
        unsigned long long o = __shfl_down(best, (unsigned)off);
        if (o > best) best = o;
      }
      if ((tid & 31) == 0) sred[tid >> 5] = best;
      __syncthreads();
      if (tid < 32) {
        best = sred[tid];
        for (int off = 16; off > 0; off >>= 1) {
          unsigned long long o = __shfl_down(best, (unsigned)off);
          if (o > best) best = o;
        }
        best = __shfl(best, 0);
        if (best == 0ull) {
          if (tid == 0) sh_done = 1;
        } else {
          int nsel = sh_nsel;
          float4 bx = boxes[key_idx(best)];
          float areaA = (bx.z - bx.x) * (bx.w - bx.y);
          bool ov = false;
          for (int q = tid; q < nsel; q += 32)
            ov = ov || (iou2(bx, areaA, selbox[q]) > IOU_THR);
          unsigned long long anyov = __ballot(ov);
          if (tid == 0) {
            if (anyov == 0ull) {
              selbox[nsel] = bx;
              out_sc[c * MAX_BOXES + nsel] = key_score(best);
              out_bx[(c * MAX_BOXES + nsel) * 4 + 0] = bx.x;
              out_bx[(c * MAX_BOXES + nsel) * 4 + 1] = bx.y;
              out_bx[(c * MAX_BOXES + nsel) * 4 + 2] = bx.z;
              out_bx[(c * MAX_BOXES + nsel) * 4 + 3] = bx.w;
              out_id[c * MAX_BOXES + nsel] = (float)(c + 1);
              sh_nsel = nsel + 1;
            }
            sh_best = best;
          }
        }
      }
    }
  }
}

extern "C" void kernel_launch(void* const* d_in, const int* in_sizes, int n_in,
                              void* d_out, int out_size, void* d_ws, size_t ws_size,
                              hipStream_t stream) {
  (void)in_sizes; (void)n_in; (void)out_size; (void)ws_size;

  const float*  conf = (const float*)d_in[0];
  const float4* anch = (const float4*)d_in[1];
  const float4* pr   = (const float4*)d_in[2];
  float* out = (float*)d_out;

  char* ws = (char*)d_ws;
  size_t off = 0;
  float4* boxes = (float4*)(ws + off);               off += (size_t)N_ANCHORS * 16;       // 1,745,920
  unsigned* ghist = (unsigned*)(ws + off);           off += (size_t)NUM_CLASSES * NBINS * 4; // 20,480
  unsigned* cnt = (unsigned*)(ws + off);             off += (size_t)NUM_CLASSES * 4;      // 320
  int* bsel = (int*)(ws + off);                      off += (size_t)NUM_CLASSES * 4;
  unsigned* totals = (unsigned*)(ws + off);          off += (size_t)NUM_CLASSES * 4;
  unsigned long long* cand = (unsigned long long*)(ws + off);
  off += (size_t)NUM_CLASSES * CAP * 8;              // 2,621,440  -> total ~4.4 MB

  // ghist and cnt are contiguous: zero both (must be re-zeroed every call)
  (void)hipMemsetAsync(ghist, 0, (size_t)(NUM_CLASSES * NBINS + NUM_CLASSES) * 4, stream);

  int blocks = (N_ANCHORS + 255) / 256;
  prep_boxes<<<blocks, 256, 0, stream>>>(anch, pr, boxes);
  hist_kernel<<<blocks, 256, 0, stream>>>(conf, ghist);
  thresh_kernel<<<1, 128, 0, stream>>>(ghist, bsel, totals);
  collect_kernel<<<blocks, 256, 0, stream>>>(conf, bsel, cnt, cand);
  nms_kernel<<<NUM_CLASSES, 1024, 0, stream>>>(conf, boxes, cnt, totals, cand, out);
}